// VoxelModule_33698313404544
// MI455X (gfx1250) — compile-verified
//
#include <hip/hip_runtime.h>
#include <hip/hip_bf16.h>
#include <stdint.h>

// Problem constants (match reference).
#define B_    16
#define N_    500000
#define CHUNK 2048
#define NBLK  ((N_ + CHUNK - 1) / CHUNK)   // 245 tiles per batch row
// Radix plan: key = vx*10000+vy*100+vz < 2^19  ->  2 stable LSD passes:
//   pass 1: digit = key & 511        (9 bits, 512 bins)
//   pass 2: digit = key >> 9  (<768) (10 bits, 768 bins)

// ---------------------------------------------------------------------------
// Kernel 0: per-point voxel key, written straight to output region 0 (which
// also serves as the pass-1 sort input).
// ---------------------------------------------------------------------------
__global__ void k_keys(const float* __restrict__ pc, const int* __restrict__ vsp,
                       int* __restrict__ keys_out) {
  int i = blockIdx.x * blockDim.x + threadIdx.x;
  if (i >= B_ * N_) return;
  int vs = *vsp;
  float m = (float)(vs - 1);
  __builtin_prefetch(pc + (size_t)i * 3 + 768, 0, 0);  // speculative global_prefetch_b8
  float x = pc[(size_t)i * 3 + 0];
  float y = pc[(size_t)i * 3 + 1];
  float z = pc[(size_t)i * 3 + 2];
  int vx = (int)(x * m);   // v_cvt_i32_f32 truncates == astype(int32) (non-negative)
  int vy = (int)(y * m);
  int vz = (int)(z * m);
  keys_out[i] = vx * 10000 + vy * 100 + vz;
}

// ---------------------------------------------------------------------------
// Per-tile digit histogram. One wave (32 threads) owns one 2048-point tile.
// ---------------------------------------------------------------------------
template <int SHIFT, int NBITS, int BINS>
__global__ void k_hist(const uint32_t* __restrict__ key_in,
                       uint32_t* __restrict__ hist) {
  __shared__ uint32_t lh[BINS];
  int lane = threadIdx.x;
  int blk  = blockIdx.x;
  int b    = blockIdx.y;
  for (int d = lane; d < BINS; d += 32) lh[d] = 0u;
  __syncthreads();
  int base = blk * CHUNK;
  for (int it = 0; it < CHUNK / 32; ++it) {
    int i = base + it * 32 + lane;
    if (i < N_) {
      __builtin_prefetch(key_in + (size_t)b * N_ + i + 256, 0, 0);
      uint32_t k = key_in[(size_t)b * N_ + i];
      atomicAdd(&lh[(k >> SHIFT) & ((1u << NBITS) - 1u)], 1u);
    }
  }
  __syncthreads();
  for (int d = lane; d < BINS; d += 32)
    hist[((size_t)b * NBLK + blk) * BINS + d] = lh[d];
}

// Barrier-free wave32 inclusive scan (5 shfl_up steps).
__device__ __forceinline__ uint32_t wave_incl_scan(uint32_t v, int lane) {
#pragma unroll
  for (int o = 1; o < 32; o <<= 1) {
    uint32_t u = (uint32_t)__shfl_up((int)v, o, 32);
    if (lane >= o) v += u;
  }
  return v;
}

// ---------------------------------------------------------------------------
// Exclusive scan over the per-batch histogram in (digit-major, tile-minor)
// order. One 256-thread workgroup per batch; 4 elements/thread + wave shuffle
// scans -> only 2 barriers per 1024-element chunk. hist is L2-resident.
// ---------------------------------------------------------------------------
template <int BINS>
__global__ void k_scan(const uint32_t* __restrict__ hist,
                       uint32_t* __restrict__ scanned) {
  __shared__ uint32_t wsum[8];
  int t    = threadIdx.x;
  int b    = blockIdx.x;
  int lane = t & 31;
  int wid  = t >> 5;
  const int L  = BINS * NBLK;
  const int CH = 1024;                 // 256 threads x 4 elements
  uint32_t carry = 0;
  for (int chunk = 0; chunk < (L + CH - 1) / CH; ++chunk) {
    int j0 = chunk * CH + t * 4;
    uint32_t e[4];
    uint32_t s = 0;
#pragma unroll
    for (int k = 0; k < 4; ++k) {      // serial 4-element pre-scan in registers
      int j = j0 + k;
      uint32_t v = 0;
      if (j < L) {
        int d   = j / NBLK;
        int blk = j % NBLK;
        v = hist[((size_t)b * NBLK + blk) * BINS + d];
      }
      e[k] = s;                        // thread-local exclusive prefix
      s += v;
    }
    uint32_t tin = wave_incl_scan(s, lane);   // inclusive over thread totals
    uint32_t tex = tin - s;                   // exclusive within wave
    if (lane == 31) wsum[wid] = tin;
    __syncthreads();
    uint32_t wex = 0, tot = 0;
#pragma unroll
    for (int w = 0; w < 8; ++w) {
      uint32_t ws = wsum[w];
      if (w < wid) wex += ws;
      tot += ws;
    }
    uint32_t base = carry + wex + tex;
#pragma unroll
    for (int k = 0; k < 4; ++k) {
      int j = j0 + k;
      if (j < L) {
        int d   = j / NBLK;
        int blk = j % NBLK;
        scanned[((size_t)b * NBLK + blk) * BINS + d] = base + e[k];
      }
    }
    carry += tot;
    __syncthreads();                   // protect wsum reuse next chunk
  }
}

// ---------------------------------------------------------------------------
// Stable in-wave rank: lanes with equal digit found with NBITS ballots;
// rank = popcount of lower same-digit lanes.
// ---------------------------------------------------------------------------
template <int NBITS>
__device__ __forceinline__ void wave_rank(bool valid, int myd, int lane,
                                          uint64_t& eq, int& rank, int& leader,
                                          uint32_t& cnt) {
  eq = __ballot(valid ? 1 : 0);
#pragma unroll
  for (int bit = 0; bit < NBITS; ++bit) {
    uint64_t bal = __ballot((valid && ((myd >> bit) & 1)) ? 1 : 0);
    eq &= ((myd >> bit) & 1) ? bal : ~bal;
  }
  uint64_t lt = (1ull << lane) - 1ull;
  rank   = (int)__popcll(eq & lt);
  leader = (int)__ffsll((unsigned long long)eq) - 1;
  if (leader < 0) leader = 0;
  cnt = (uint32_t)__popcll(eq);
}

// Async-stage 1024 u32 (4 KB) from global into LDS: 8 x b128 issues,
// 512 B per issue (32 lanes x 16 B). LDS byte offset = low 32 bits of the
// generic shared-memory address (ISA 10.2).
__device__ __forceinline__ void async_stage_half(const uint32_t* __restrict__ gsrc,
                                                 uint32_t* ldst, int lane) {
  unsigned lbase = (unsigned)(size_t)ldst;
#pragma unroll
  for (int j = 0; j < 8; ++j) {
    unsigned loffb = lbase + (unsigned)(j * 512 + lane * 16);
    const uint32_t* gp = gsrc + j * 128 + lane * 4;
    asm volatile("global_load_async_to_lds_b128 %0, %1, off"
                 :: "v"(loffb), "v"(gp) : "memory");
  }
}

// ---------------------------------------------------------------------------
// Stable scatter pass. One wave per 2048-key tile. The tile is async-staged
// into LDS in two 4 KB halves (second half's DMA overlaps first half's
// rank/scatter chain). The running bin offsets live in LDS and advance with
// ds_add_rtn_u32 - DS ops from one wave are architecturally in-order, so
// iteration order == original index order => stability.
// FINAL=false: pass 1 (payload idx is implicit = i, outputs to ping buffers).
// FINAL=true : pass 2 (reads idx payload, writes order + sorted_keys).
// ---------------------------------------------------------------------------
template <int SHIFT, int NBITS, int BINS, bool FINAL>
__global__ void k_scatter(const uint32_t* __restrict__ key_in,
                          const int* __restrict__ idx_in,
                          const uint32_t* __restrict__ scanned,
                          uint32_t* __restrict__ key_out,
                          int* __restrict__ idx_out) {
  __shared__ uint32_t loff[BINS];
  __shared__ uint32_t stage[CHUNK];
  int lane = threadIdx.x;
  int blk  = blockIdx.x;
  int b    = blockIdx.y;

  // Kick off the tile DMA first so it overlaps the offset-table load.
  const uint32_t* src = key_in + (size_t)b * N_ + blk * CHUNK;
  async_stage_half(src,        &stage[0],    lane);
  async_stage_half(src + 1024, &stage[1024], lane);

  const uint32_t* sc = scanned + ((size_t)b * NBLK + blk) * BINS;
  for (int d = lane; d < BINS; d += 32) loff[d] = sc[d];
  __syncthreads();

  for (int half = 0; half < 2; ++half) {
    if (half == 0)
      asm volatile("s_wait_asynccnt 0x8" ::: "memory");  // first 4KB landed
    else
      asm volatile("s_wait_asynccnt 0x0" ::: "memory");  // second 4KB landed
    for (int it = 0; it < 32; ++it) {
      int local = half * 1024 + it * 32 + lane;
      int i = blk * CHUNK + local;
      bool valid = i < N_;
      uint32_t key = stage[local];
      int myd = valid ? (int)((key >> SHIFT) & ((1u << NBITS) - 1u)) : -1;
      uint64_t eq; int rank, leader; uint32_t cnt;
      wave_rank<NBITS>(valid, myd, lane, eq, rank, leader, cnt);
      uint32_t basev = 0;
      if (valid && lane == leader)
        basev = atomicAdd(&loff[myd], cnt);               // ds_add_rtn, in-order
      basev = (uint32_t)__shfl((int)basev, leader, 32);
      if (valid) {
        uint32_t pos = basev + (uint32_t)rank;
        key_out[(size_t)b * N_ + pos] = key;              // pass2: sorted_keys
        int iv;
        if constexpr (FINAL) iv = idx_in[(size_t)b * N_ + i];
        else                 iv = i;
        idx_out[(size_t)b * N_ + pos] = iv;               // pass2: order
      }
    }
  }
}

// ---------------------------------------------------------------------------
extern "C" void kernel_launch(void* const* d_in, const int* in_sizes, int n_in,
                              void* d_out, int out_size, void* d_ws, size_t ws_size,
                              hipStream_t stream) {
  (void)in_sizes; (void)n_in; (void)out_size; (void)ws_size;
  const float* pc  = (const float*)d_in[0];
  const int*   vsp = (const int*)d_in[1];

  int* keys_out  = (int*)d_out;                      // [B,N] keys (also pass-1 input)
  int* order_out = keys_out + (size_t)B_ * N_;       // [B,N] order
  int* skeys_out = order_out + (size_t)B_ * N_;      // [B,N] sorted_keys

  char* ws = (char*)d_ws;
  size_t off = 0;
  auto carve = [&](size_t bytes) {
    void* p = ws + off;
    off = (off + bytes + 255) & ~(size_t)255;
    return p;
  };
  uint32_t* key1 = (uint32_t*)carve((size_t)B_ * N_ * 4);          // 32 MB ping keys
  int*      idx1 = (int*)carve((size_t)B_ * N_ * 4);               // 32 MB ping idx
  uint32_t* hist = (uint32_t*)carve((size_t)B_ * NBLK * 768 * 4);  // ~12 MB (max bins)
  uint32_t* scn  = (uint32_t*)carve((size_t)B_ * NBLK * 768 * 4);  // ~12 MB

  int total = B_ * N_;
  k_keys<<<(total + 255) / 256, 256, 0, stream>>>(pc, vsp, keys_out);

  dim3 tiles(NBLK, B_);
  // Pass 1: digit = key & 511
  k_hist<0, 9, 512><<<tiles, 32, 0, stream>>>((const uint32_t*)keys_out, hist);
  k_scan<512><<<B_, 256, 0, stream>>>(hist, scn);
  k_scatter<0, 9, 512, false><<<tiles, 32, 0, stream>>>(
      (const uint32_t*)keys_out, nullptr, scn, key1, idx1);
  // Pass 2: digit = key >> 9 (< 768)
  k_hist<9, 10, 768><<<tiles, 32, 0, stream>>>(key1, hist);
  k_scan<768><<<B_, 256, 0, stream>>>(hist, scn);
  k_scatter<9, 10, 768, true><<<tiles, 32, 0, stream>>>(
      key1, idx1, scn, (uint32_t*)skeys_out, order_out);
}